// NDC_70841190580829
// MI455X (gfx1250) — compile-verified
//
#include <hip/hip_runtime.h>

// ---------------------------------------------------------------------------
// WDF diode-clipper scan for MI455X (gfx1250, wave32).
//
// ndc_prep: parallel precompute of per-timestep WDF coefficients into d_ws:
//   8 floats/step = {v, row0[0..2], row1[0..2], dp_r}; padded T+8 steps so the
//   scan kernel needs no index clamping (t+2 loads, t+8 prefetch).
//
// ndc_scan: ONE wave (32 lanes) runs the strictly sequential 480k-step
//   recurrence. 32x32 hidden layers via V_WMMA_F32_16X16X4_F32:
//     D = A x B, A(16x4)[m,k] = h[4c+k]  (ds_bpermute gathers, all 16 issued
//     before the WMMA chain so DS latency is paid once),
//     B(4x16)[k,n] = W[n,4c+k]           (W^T tiles resident in VGPRs).
//   Per N-block the 8 K-chunks form TWO independent 4-deep WMMA chains
//   (chunks 0-3 / 4-7), joined by one v_add on D-element 0 — this sizing
//   (8 v8f accumulators total) is what the RA schedules cleanly; 16 chains
//   caused accumulator-range overlap and WAW serialization.
//   Chunk 0-3 gathers read only source lanes 0-15 and chunk 4-7 gathers only
//   lanes 16-31, so no merge cndmask between layers: hLo/hHi relu outputs
//   directly feed the two gather halves.
//   Output dot needs no selects: p = WoA*gLo + WoB*gHi with
//   WoA[lane]=W_out[lane&15], WoB[lane]=W_out[16+(lane&15)]; p depends only
//   on lane&15, so 4 DPP row_ror adds complete the 32-element reduction.
//   Per-step params are software-pipelined TWO steps deep (scalar-path loads).
// ---------------------------------------------------------------------------

typedef __attribute__((ext_vector_type(2))) float v2f;
typedef __attribute__((ext_vector_type(8))) float v8f;

#define C1F 4.7e-9f

#if __has_builtin(__builtin_amdgcn_update_dpp)
#define ROR_ADD(x, ctrl)                                                  \
  ((x) + __int_as_float(__builtin_amdgcn_update_dpp(                      \
             0, __float_as_int(x), (ctrl), 0xf, 0xf, true)))
#define HAVE_DPP 1
#endif

__global__ void ndc_prep(const float* __restrict__ v_in,
                         const float* __restrict__ vs_r,
                         const float* __restrict__ fs,
                         float* __restrict__ params, int T) {
  int t = blockIdx.x * blockDim.x + threadIdx.x;
  if (t >= T + 8) return;            // padding for the scan's lookahead
  int ti = (t < T) ? t : (T - 1);
  float f   = fs[ti];
  float vr  = vs_r[ti];
  float c1r = 1.0f / (2.0f * C1F * f);
  float r0  = c1r * vr / (c1r + vr);
  float g0 = 1.0f / r0, g1 = 1.0f / c1r, g2 = 1.0f / vr;
  float inv = 2.0f / (g0 + g1 + g2);
  float P0 = g0 * inv, P1 = g1 * inv, P2 = g2 * inv;
  float4* p4 = (float4*)(params + 8 * (size_t)t);
  p4[0] = make_float4(v_in[ti], P0 - 1.0f, P1, P2);
  p4[1] = make_float4(P0, P1 - 1.0f, P2, r0 * (1.0f / 3000.0f));
}

__global__ __launch_bounds__(32, 1) void ndc_scan(
    const float* __restrict__ params,
    const float* __restrict__ W_in, const float* __restrict__ b_in,
    const float* __restrict__ W_h,  const float* __restrict__ b_h,
    const float* __restrict__ W_out, const float* __restrict__ b_out,
    float* __restrict__ out, int T) {
  const int  lane = threadIdx.x;          // 0..31, one wave
  const bool hi   = lane >= 16;
  const int  r16  = lane & 15;

  // ---- per-lane small weights --------------------------------------------
  float Wi0 = W_in[2 * lane];
  float Wi1 = W_in[2 * lane + 1];
  float bin = b_in[lane];
  // bias views: bLo*[lane] = b[l][lane&15], bHi*[lane] = b[l][16+(lane&15)]
  float bLo0 = b_h[r16];
  float bHi0 = b_h[16 + r16];
  float bLo1 = b_h[32 + r16];
  float bHi1 = b_h[48 + r16];
  float WoA = W_out[r16];        // pairs with gLo
  float WoB = W_out[16 + r16];   // pairs with gHi
  float bo  = b_out[0];

  // ---- B-operand tiles (W^T) for the two 32x32 hidden layers -------------
  //   VGPR0: lanes 0-15 = B[K=0, N=lane], lanes 16-31 = B[K=2, N=lane-16]
  //   VGPR1: lanes 0-15 = B[K=1, N=lane], lanes 16-31 = B[K=3, N=lane-16]
  // B[k,n] = W[16*blk + n, 4c + k].
  float Bw[2][8][2][2];
#pragma unroll
  for (int l = 0; l < 2; ++l)
#pragma unroll
    for (int c = 0; c < 8; ++c)
#pragma unroll
      for (int blk = 0; blk < 2; ++blk) {
        int row = 16 * blk + r16;
        int col = 4 * c + (hi ? 2 : 0);
        Bw[l][c][blk][0] = W_h[l * 1024 + row * 32 + col];
        Bw[l][c][blk][1] = W_h[l * 1024 + row * 32 + col + 1];
      }

  // A-operand gather base (bytes): lanes 0-15 pick h[4c], 16-31 pick h[4c+2]
  const int idx_base = hi ? 8 : 0;

  const float4* pf = (const float4*)params;
  float4 qa   = pf[0], qb   = pf[1];   // step t
  float4 qa_n = pf[2], qb_n = pf[3];   // step t+1

  float a0 = 0.0f, b1 = 0.0f;

  for (int t = 0; t < T; ++t) {
    // --- 2-deep pipeline: issue loads for step t+2 now (buffer padded) ----
    float4 qa_nn = pf[2 * (t + 2)];
    float4 qb_nn = pf[2 * (t + 2) + 1];
    __builtin_prefetch(params + 8 * (size_t)(t + 8), 0, 3);

    float v   = qa.x;
    float r00 = qa.y, r01 = qa.z, r02 = qa.w;
    float r10 = qb.x, r11 = qb.y, r12 = qb.z;
    float drt = qb.w;

    // dp_a = row0 . [a0, b1, v]   (lane-uniform)
    float dp_a = r00 * a0 + r01 * b1 + r02 * v;

    // input layer 2 -> 32  (lane i owns h[i], all 32 lanes valid)
    float h0 = fmaxf(Wi0 * dp_a + Wi1 * drt + bin, 0.0f);
    int hb = __float_as_int(h0);

    // ---------------- hidden layer 1 ----------------
    int Ax[8], Ay[8];
#pragma unroll
    for (int c = 0; c < 8; ++c) {
      Ax[c] = __builtin_amdgcn_ds_bpermute(16 * c + idx_base, hb);
      Ay[c] = __builtin_amdgcn_ds_bpermute(16 * c + idx_base + 4, hb);
    }
    v8f a00 = {}, a01 = {}, a10 = {}, a11 = {};
#pragma unroll
    for (int c = 0; c < 4; ++c) {
      v2f A;  A.x  = __int_as_float(Ax[c]);     A.y  = __int_as_float(Ay[c]);
      v2f A2; A2.x = __int_as_float(Ax[c + 4]); A2.y = __int_as_float(Ay[c + 4]);
      v2f B00; B00.x = Bw[0][c][0][0];     B00.y = Bw[0][c][0][1];
      v2f B01; B01.x = Bw[0][c][1][0];     B01.y = Bw[0][c][1][1];
      v2f B10; B10.x = Bw[0][c + 4][0][0]; B10.y = Bw[0][c + 4][0][1];
      v2f B11; B11.x = Bw[0][c + 4][1][0]; B11.y = Bw[0][c + 4][1][1];
      a00 = __builtin_amdgcn_wmma_f32_16x16x4_f32(false, A,  false, B00,
                                                  (short)0, a00, false, false);
      a10 = __builtin_amdgcn_wmma_f32_16x16x4_f32(false, A,  false, B01,
                                                  (short)0, a10, false, false);
      a01 = __builtin_amdgcn_wmma_f32_16x16x4_f32(false, A2, false, B10,
                                                  (short)0, a01, false, false);
      a11 = __builtin_amdgcn_wmma_f32_16x16x4_f32(false, A2, false, B11,
                                                  (short)0, a11, false, false);
    }
    // block0 D0: lanes0-15 = y[lane], lanes16-31 = y[lane-16]
    // block1 D0: lanes0-15 = y[16+lane], lanes16-31 = y[lane]
    // hLo: lanes 0-15 hold h[0..15]; hHi: lanes 16-31 hold h[16..31]
    float hLo = fmaxf((a00[0] + a01[0]) + bLo0, 0.0f);
    float hHi = fmaxf((a10[0] + a11[0]) + bHi0, 0.0f);
    int hbL = __float_as_int(hLo);
    int hbH = __float_as_int(hHi);

    // ---------------- hidden layer 2 ----------------
    // chunks 0-3 gather bytes 0..63   -> source lanes 0-15  (hLo valid there)
    // chunks 4-7 gather bytes 64..124 -> source lanes 16-31 (hHi valid there)
#pragma unroll
    for (int c = 0; c < 4; ++c) {
      Ax[c] = __builtin_amdgcn_ds_bpermute(16 * c + idx_base, hbL);
      Ay[c] = __builtin_amdgcn_ds_bpermute(16 * c + idx_base + 4, hbL);
    }
#pragma unroll
    for (int c = 4; c < 8; ++c) {
      Ax[c] = __builtin_amdgcn_ds_bpermute(16 * c + idx_base, hbH);
      Ay[c] = __builtin_amdgcn_ds_bpermute(16 * c + idx_base + 4, hbH);
    }
    v8f b00 = {}, b01 = {}, b10 = {}, b11 = {};
#pragma unroll
    for (int c = 0; c < 4; ++c) {
      v2f A;  A.x  = __int_as_float(Ax[c]);     A.y  = __int_as_float(Ay[c]);
      v2f A2; A2.x = __int_as_float(Ax[c + 4]); A2.y = __int_as_float(Ay[c + 4]);
      v2f B00; B00.x = Bw[1][c][0][0];     B00.y = Bw[1][c][0][1];
      v2f B01; B01.x = Bw[1][c][1][0];     B01.y = Bw[1][c][1][1];
      v2f B10; B10.x = Bw[1][c + 4][0][0]; B10.y = Bw[1][c + 4][0][1];
      v2f B11; B11.x = Bw[1][c + 4][1][0]; B11.y = Bw[1][c + 4][1][1];
      b00 = __builtin_amdgcn_wmma_f32_16x16x4_f32(false, A,  false, B00,
                                                  (short)0, b00, false, false);
      b10 = __builtin_amdgcn_wmma_f32_16x16x4_f32(false, A,  false, B01,
                                                  (short)0, b10, false, false);
      b01 = __builtin_amdgcn_wmma_f32_16x16x4_f32(false, A2, false, B10,
                                                  (short)0, b01, false, false);
      b11 = __builtin_amdgcn_wmma_f32_16x16x4_f32(false, A2, false, B11,
                                                  (short)0, b11, false, false);
    }
    float gLo = fmaxf((b00[0] + b01[0]) + bLo1, 0.0f);  // h2[lane&15]
    float gHi = fmaxf((b10[0] + b11[0]) + bHi1, 0.0f);  // h2[16+(lane&15)]

    // output layer 32 -> 1: p depends only on lane&15 (rows identical), so
    // 4 DPP row_ror adds give the full 32-element sum in every lane.
    float p = WoA * gLo + WoB * gHi;
#ifdef HAVE_DPP
    p = ROR_ADD(p, 0x128);  // row_ror:8
    p = ROR_ADD(p, 0x124);  // row_ror:4
    p = ROR_ADD(p, 0x122);  // row_ror:2
    p = ROR_ADD(p, 0x121);  // row_ror:1
#else
#pragma unroll
    for (int m = 8; m >= 1; m >>= 1) {
      p += __int_as_float(
          __builtin_amdgcn_ds_bpermute((lane ^ m) << 2, __float_as_int(p)));
    }
#endif
    float dp_b = p + bo;  // lane-uniform

    // b1n = row1 . [dp_b, b1, v]
    float b1n = r10 * dp_b + r11 * b1 + r12 * v;

    // dp_a, dp_b are lane-uniform: all 32 lanes store the same value to the
    // same address (well-defined, avoids an exec-mask save/restore per step)
    out[t] = 0.5f * (dp_a + dp_b);

    a0 = dp_b;
    b1 = b1n;
    qa = qa_n;  qb = qb_n;
    qa_n = qa_nn;  qb_n = qb_nn;
  }
}

extern "C" void kernel_launch(void* const* d_in, const int* in_sizes, int n_in,
                              void* d_out, int out_size, void* d_ws,
                              size_t ws_size, hipStream_t stream) {
  const float* v_in  = (const float*)d_in[0];
  const float* vs_r  = (const float*)d_in[1];
  const float* fs    = (const float*)d_in[2];
  const float* W_in  = (const float*)d_in[3];
  const float* b_in  = (const float*)d_in[4];
  const float* W_h   = (const float*)d_in[5];
  const float* b_h   = (const float*)d_in[6];
  const float* W_out = (const float*)d_in[7];
  const float* b_out = (const float*)d_in[8];
  float* out = (float*)d_out;
  int T = in_sizes[0];

  float* params = (float*)d_ws;  // 8 floats per step, T+8 steps

  ndc_prep<<<(T + 8 + 255) / 256, 256, 0, stream>>>(v_in, vs_r, fs, params, T);
  ndc_scan<<<1, 32, 0, stream>>>(params, W_in, b_in, W_h, b_h, W_out, b_out,
                                 out, T);
}